// SelfSimilarity_91173565759903
// MI455X (gfx1250) — compile-verified
//
#include <hip/hip_runtime.h>
#include <hip/hip_bf16.h>

typedef float v2f __attribute__((ext_vector_type(2)));
typedef float v8f __attribute__((ext_vector_type(8)));

#define BATCH   4
#define CHAN    128
#define HH      128
#define WW      128
#define IMG     (HH*WW)          // 16384 floats per (b,c) image
#define NP      65               // p = 0..64
#define NQ      129              // q index 0..128  (q-64 in [-64,64])
#define KC      24               // K-rows per LDS stage buffer
#define LSTR    26               // padded LDS row stride (even -> 8B aligned v2f)
#define NSTAGE  64
#define CK      (KC*NSTAGE)      // 1536 K-rows per workgroup chunk
#define MAXCHUNK 43              // ceil(512*128 / 1536)

#if defined(__HIP_DEVICE_COMPILE__) && __has_builtin(__builtin_amdgcn_global_load_async_to_lds_b32)
#define HAS_ASYNC_LDS 1
#else
#define HAS_ASYNC_LDS 0
#endif

#if defined(__HIP_DEVICE_COMPILE__) && __has_builtin(__builtin_amdgcn_s_wait_asynccnt)
#define WAIT_ASYNC0() __builtin_amdgcn_s_wait_asynccnt(0)
#elif HAS_ASYNC_LDS
#define WAIT_ASYNC0() asm volatile("s_wait_asynccnt 0x0" ::: "memory")
#else
#define WAIT_ASYNC0() ((void)0)
#endif

// one f32 element: global -> LDS (async DMA-style if available, sync otherwise)
__device__ __forceinline__ void stage_f32(const float* g, float* l) {
#if HAS_ASYNC_LDS
    __builtin_amdgcn_global_load_async_to_lds_b32(
        (__attribute__((address_space(1))) int*)(const_cast<float*>(g)),
        (__attribute__((address_space(3))) int*)(l),
        0, 0);
#else
    *l = *g;
#endif
}

// ---------------------------------------------------------------------------
// Pass A: S[i,j] = sum_{b,c} x^2 ; Mean[b,i,j] = mean_c x
// ---------------------------------------------------------------------------
__global__ void moments_kernel(const float* __restrict__ x,
                               float* __restrict__ S,
                               float* __restrict__ Mean) {
    int idx = blockIdx.x * blockDim.x + threadIdx.x;   // pixel (i,j)
    float s = 0.f, m0 = 0.f, m1 = 0.f, m2 = 0.f, m3 = 0.f;
    for (int c = 0; c < CHAN; ++c) {
        float v0 = x[(0*CHAN + c)*IMG + idx];
        float v1 = x[(1*CHAN + c)*IMG + idx];
        float v2 = x[(2*CHAN + c)*IMG + idx];
        float v3 = x[(3*CHAN + c)*IMG + idx];
        s  += v0*v0 + v1*v1 + v2*v2 + v3*v3;
        m0 += v0; m1 += v1; m2 += v2; m3 += v3;
    }
    S[idx] = s;
    const float inv = 1.0f / (float)CHAN;
    Mean[0*IMG + idx] = m0 * inv;
    Mean[1*IMG + idx] = m1 * inv;
    Mean[2*IMG + idx] = m2 * inv;
    Mean[3*IMG + idx] = m3 * inv;
}

// ---------------------------------------------------------------------------
// Pass B: integral image I[129][129] of S
// ---------------------------------------------------------------------------
__global__ void integral_kernel(const float* __restrict__ S,
                                float* __restrict__ I) {
    int t = threadIdx.x;
    if (t < NQ) I[t] = 0.f;            // row 0
    if (t < NQ) I[t * NQ] = 0.f;       // col 0
    if (t >= 1 && t <= HH) {           // row cumsum
        int i = t - 1;
        float run = 0.f;
        for (int j = 0; j < WW; ++j) {
            run += S[i*WW + j];
            I[t*NQ + (j+1)] = run;
        }
    }
    __syncthreads();
    if (t >= 1 && t <= WW) {           // column cumsum
        float run = 0.f;
        for (int i = 1; i <= HH; ++i) {
            run += I[i*NQ + t];
            I[i*NQ + t] = run;
        }
    }
}

// ---------------------------------------------------------------------------
// Pass C: split-K fp32 WMMA GEMM with double-buffered async LDS staging.
//   G_p[j][j'] = sum_r A_p[r][j]*B_p[r][j'];  Xacc[p][64+d] += diag_d(G_p)
// ---------------------------------------------------------------------------
__global__ void __launch_bounds__(256)
xcorr_gemm_kernel(const float* __restrict__ x, float* __restrict__ Xacc) {
    const int p = blockIdx.y;
    const unsigned rows = (unsigned)(HH - p);
    const unsigned Kp   = (unsigned)(BATCH*CHAN) * rows;
    const unsigned k0   = blockIdx.x * (unsigned)CK;
    if (k0 >= Kp) return;

    __shared__ float lA[2 * WW * LSTR];   // lA[buf][j*LSTR + rr]
    __shared__ float lB[2 * WW * LSTR];
    __shared__ float diag[8 * 132];

    const int wave = threadIdx.x >> 5;
    const int lane = threadIdx.x & 31;
    const int m    = lane & 15;
    const int ksel = (lane >> 4) << 1;    // 0 (lanes 0-15) / 2 (lanes 16-31)

    const unsigned krem = Kp - k0;
    const unsigned ktot = krem < (unsigned)CK ? krem : (unsigned)CK;
    const int nst = (int)((ktot + KC - 1) / KC);

    // stage one KC-row chunk of A_p and B_p into buffer `buf`
    auto issue_stage = [&](unsigned kbase, int buf) {
        float* bufA = lA + buf * (WW * LSTR);
        float* bufB = lB + buf * (WW * LSTR);
        for (int t = threadIdx.x; t < KC * WW; t += 256) {
            int rr = t >> 7;
            int j  = t & (WW - 1);
            unsigned r = kbase + (unsigned)rr;
            int laddr = j * LSTR + rr;
            if (r < Kp) {
                unsigned bc = r / rows;
                unsigned i  = r - bc * rows;
                const float* img = x + bc * IMG;
                stage_f32(&img[i * WW + j],       &bufA[laddr]);
                stage_f32(&img[(i + p) * WW + j], &bufB[laddr]);
            } else {
                bufA[laddr] = 0.f;
                bufB[laddr] = 0.f;
            }
        }
    };

    v8f acc[8];
#pragma unroll
    for (int nt = 0; nt < 8; ++nt) acc[nt] = (v8f)(0.f);

    issue_stage(k0, 0);

    for (int s = 0; s < nst; ++s) {
        WAIT_ASYNC0();          // my async writes into buffer s&1 have landed
        __syncthreads();        // everyone's writes visible; prev buffer free
        if (s + 1 < nst) issue_stage(k0 + (unsigned)((s + 1) * KC), (s + 1) & 1);

        const float* lAc = lA + (s & 1) * (WW * LSTR);
        const float* lBc = lB + (s & 1) * (WW * LSTR);
#pragma unroll
        for (int kk = 0; kk < KC; kk += 4) {
            v2f a = *(const v2f*)&lAc[(16 * wave + m) * LSTR + kk + ksel];
            v2f b[8];
#pragma unroll
            for (int nt = 0; nt < 8; ++nt)
                b[nt] = *(const v2f*)&lBc[(16 * nt + m) * LSTR + kk + ksel];
#pragma unroll
            for (int nt = 0; nt < 8; ++nt)
                acc[nt] = __builtin_amdgcn_wmma_f32_16x16x4_f32(
                    false, a, false, b[nt], (short)0, acc[nt], false, false);
        }
    }
    __syncthreads();

    // ---- epilogue: diagonal reduction (LDS atomics, then global atomics) ----
    for (int t = threadIdx.x; t < 8 * 132; t += 256) diag[t] = 0.f;
    __syncthreads();

    float* mydiag = &diag[wave * 132];
    const int mhalf = (lane < 16) ? 0 : 8;
#pragma unroll
    for (int nt = 0; nt < 8; ++nt) {
#pragma unroll
        for (int v = 0; v < 8; ++v) {
            int j  = 16 * wave + v + mhalf;   // output row (C-layout VGPR v)
            int jp = 16 * nt + m;             // output col
            int d  = jp - j;
            if (d >= -64 && d <= 64)
                atomicAdd(&mydiag[64 + d], acc[nt][v]);
        }
    }
    __syncthreads();

    for (int t = threadIdx.x; t < NQ; t += 256) {
        float sum = 0.f;
#pragma unroll
        for (int w2 = 0; w2 < 8; ++w2) sum += diag[w2 * 132 + t];
        atomicAdd(&Xacc[p * NQ + t], sum);
    }
}

// ---------------------------------------------------------------------------
// Pass D: scores + gather.  out[b,0,k,l] = score(k,l) * Mean[b, k%128, l%128]
// ---------------------------------------------------------------------------
__global__ void finalize_kernel(const float* __restrict__ Mean,
                                const float* __restrict__ I,
                                const float* __restrict__ Xacc,
                                float* __restrict__ out) {
    int idx = blockIdx.x * blockDim.x + threadIdx.x;
    if (idx >= BATCH * NQ * NQ) return;
    int l = idx % NQ;
    int t = idx / NQ;
    int k = t % NQ;
    int b = t / NQ;

    const float total = I[HH * NQ + WW];
    const float denom = sqrtf(total);
    const float eps   = 1e-10f * total;

    int p = k - 64, q = l - 64;

    auto rect = [&](int pp, int qq) -> float {
        int r0 = max(pp, 0), r1 = HH + min(pp, 0);
        int c0 = max(qq, 0), c1 = WW + min(qq, 0);
        return I[r1*NQ + c1] - I[r0*NQ + c1] - I[r1*NQ + c0] + I[r0*NQ + c0];
    };
    float sumA = rect(p, q);
    float sumB = rect(-p, -q);

    float X = (p >= 0) ? Xacc[p * NQ + (64 + q)]
                       : Xacc[(-p) * NQ + (64 - q)];

    float num_sq = fmaxf(sumA + sumB - 2.0f * X, 0.0f);
    float numer  = (num_sq > eps) ? sqrtf(num_sq) : 0.0f;
    float score  = -(numer / denom);

    int ii = k & (HH - 1);
    int jj = l & (WW - 1);
    out[idx] = score * Mean[b * IMG + ii * WW + jj];
}

// ---------------------------------------------------------------------------
extern "C" void kernel_launch(void* const* d_in, const int* in_sizes, int n_in,
                              void* d_out, int out_size, void* d_ws, size_t ws_size,
                              hipStream_t stream) {
    const float* x = (const float*)d_in[0];
    float* out = (float*)d_out;
    float* ws  = (float*)d_ws;

    float* S    = ws;                         // 16384
    float* Mean = ws + 16384;                 // 65536
    float* I    = ws + 16384 + 65536;         // 16641
    float* Xacc = ws + 16384 + 65536 + 16641; // 65*129 = 8385

    (void)hipMemsetAsync(Xacc, 0, NP * NQ * sizeof(float), stream);

    moments_kernel<<<IMG / 256, 256, 0, stream>>>(x, S, Mean);
    integral_kernel<<<1, 160, 0, stream>>>(S, I);
    xcorr_gemm_kernel<<<dim3(MAXCHUNK, NP), 256, 0, stream>>>(x, Xacc);

    int nout = BATCH * NQ * NQ;
    finalize_kernel<<<(nout + 255) / 256, 256, 0, stream>>>(Mean, I, Xacc, out);
}